// Weightfun_78374563217418
// MI455X (gfx1250) — compile-verified
//
#include <hip/hip_runtime.h>
#include <hip/hip_bf16.h>
#include <stdint.h>

// weight_i = 0 if loss_i > eta else 1 - loss_i/eta ; eta = clamp(eta_value, ln(10), 20)
// Memory-bound streaming map: 128 MB in + 128 MB out => ~11 us at 23.3 TB/s.
// CDNA5 path: double-buffered GLOBAL_LOAD_ASYNC_TO_LDS_B128 pipeline (ASYNCcnt),
// ds_load_b128 readback, non-temporal global_store_b128 for the output stream.

typedef float v4f __attribute__((ext_vector_type(4)));

#define ETA_MIN 2.302585092994046f /* ln(10) */
#define ETA_MAX 20.0f
#define THREADS 256

// Async 16B copy global -> LDS (per-lane). VDST = LDS byte offset, VADDR = 64-bit
// global address, saddr = off (GV mode). Tracked by ASYNCcnt.
__device__ __forceinline__ void async_load16(const v4f* __restrict__ gsrc,
                                             unsigned lds_off) {
    asm volatile("global_load_async_to_lds_b128 %0, %1, off"
                 :
                 : "v"(lds_off), "v"(gsrc)
                 : "memory");
}

__device__ __forceinline__ void wait_async_le1() {
#if __has_builtin(__builtin_amdgcn_s_wait_asynccnt)
    __builtin_amdgcn_s_wait_asynccnt(1);
#else
    asm volatile("s_wait_asynccnt 0x1" ::: "memory");
#endif
}

__device__ __forceinline__ void wait_async_0() {
#if __has_builtin(__builtin_amdgcn_s_wait_asynccnt)
    __builtin_amdgcn_s_wait_asynccnt(0);
#else
    asm volatile("s_wait_asynccnt 0x0" ::: "memory");
#endif
}

__global__ __launch_bounds__(THREADS) void Weightfun_78374563217418_kernel(
    const float* __restrict__ loss,
    const float* __restrict__ eta_value,
    float* __restrict__ out,
    int nv4)
{
    // Double-buffered per-lane 16B slots: 2 * 256 * 16B = 8 KB LDS.
    __shared__ v4f smem[2][THREADS];

    // Uniform scalar: SMEM load, clamped once; reciprocal hoisted (one IEEE div).
    const float eta     = fminf(fmaxf(eta_value[0], ETA_MIN), ETA_MAX);
    const float inv_eta = 1.0f / eta;

    const int stride = gridDim.x * blockDim.x;
    int i = blockIdx.x * blockDim.x + threadIdx.x;

    // Low 32 bits of a flat shared pointer == LDS byte offset (aperture rule).
    const unsigned slot0 = (unsigned)(uintptr_t)&smem[0][threadIdx.x];
    const unsigned slot1 = (unsigned)(uintptr_t)&smem[1][threadIdx.x];

    if (i < nv4) async_load16((const v4f*)loss + i, slot0);

    int cur = 0;
    for (; i < nv4; i += stride) {
        const int  nxt       = i + stride;
        const bool have_next = (nxt < nv4);

        // Prefetch next tile into the other buffer while we consume this one.
        if (have_next) async_load16((const v4f*)loss + nxt, cur ? slot0 : slot1);

        // Async loads complete in order: <=1 outstanding guarantees the current
        // buffer's copy has landed in LDS. Wave-local slots => no barrier needed.
        if (have_next) wait_async_le1();
        else           wait_async_0();

        v4f l = smem[cur][threadIdx.x];   // ds_load_b128, bank-conflict-free
        v4f w;
#pragma unroll
        for (int k = 0; k < 4; ++k) {
            float li = l[k];
            float v  = fmaf(-li, inv_eta, 1.0f);  // 1 - li/eta
            w[k]     = (li > eta) ? 0.0f : v;
        }
        // Output stream bypasses cache retention: global_store_b128 th:NT.
        __builtin_nontemporal_store(w, (v4f*)out + i);
        cur ^= 1;
    }
}

// Generic tail for N % 4 != 0 (not hit for N = 2^25, kept for safety).
__global__ void Weightfun_78374563217418_tail(
    const float* __restrict__ loss,
    const float* __restrict__ eta_value,
    float* __restrict__ out,
    int start, int n)
{
    int i = start + blockIdx.x * blockDim.x + threadIdx.x;
    if (i < n) {
        const float eta     = fminf(fmaxf(eta_value[0], ETA_MIN), ETA_MAX);
        const float inv_eta = 1.0f / eta;
        float li = loss[i];
        float v  = fmaf(-li, inv_eta, 1.0f);
        out[i]   = (li > eta) ? 0.0f : v;
    }
}

extern "C" void kernel_launch(void* const* d_in, const int* in_sizes, int n_in,
                              void* d_out, int out_size, void* d_ws, size_t ws_size,
                              hipStream_t stream) {
    (void)n_in; (void)out_size; (void)d_ws; (void)ws_size;

    const float* loss = (const float*)d_in[0];
    const float* eta  = (const float*)d_in[1];
    float*       out  = (float*)d_out;

    const int n    = in_sizes[0];
    const int nv4  = n >> 2;          // float4 count
    const int tail = n - (nv4 << 2);

    if (nv4 > 0) {
        const int threads = THREADS;  // 8 wave32s per block
        // ~4 vec4 (16 floats) per thread: enough waves to flood every WGP while
        // keeping the async pipeline's trip count short.
        long long want_threads = (nv4 + 3) / 4;
        int blocks = (int)((want_threads + threads - 1) / threads);
        if (blocks < 1)     blocks = 1;
        if (blocks > 65535) blocks = 65535;
        Weightfun_78374563217418_kernel<<<blocks, threads, 0, stream>>>(loss, eta, out, nv4);
    }
    if (tail > 0) {
        Weightfun_78374563217418_tail<<<1, 32, 0, stream>>>(loss, eta, out, nv4 << 2, n);
    }
}